// GAT_58136677318878
// MI455X (gfx1250) — compile-verified
//
#include <hip/hip_runtime.h>
#include <hip/hip_bf16.h>
#include <math.h>

typedef __attribute__((ext_vector_type(2))) float v2f;
typedef __attribute__((ext_vector_type(8))) float v8f;

#define NEG_SLOPE 0.2f

// ---------- order-preserving float <-> uint map for atomic max ----------
__device__ __forceinline__ unsigned flipF(float f) {
    unsigned u = __float_as_uint(f);
    return (u & 0x80000000u) ? ~u : (u | 0x80000000u);
}
__device__ __forceinline__ float unflipF(unsigned u) {
    unsigned v = (u & 0x80000000u) ? (u & 0x7FFFFFFFu) : ~u;
    return __uint_as_float(v);
}

// ---------- zero scratch accumulators ----------
__global__ void k_zero(float* __restrict__ p, size_t count) {
    size_t i = (size_t)blockIdx.x * blockDim.x + threadIdx.x;
    if (i < count) p[i] = 0.0f;
}

// ---------- GEMM1: h1[N,32] = x[N,128] @ W1[128,32] via V_WMMA_F32_16X16X4_F32 ----------
// block = 256 threads = 8 waves; each wave -> 16 rows x 32 cols; block covers 128 rows.
__global__ void k_gemm1_wmma(const float* __restrict__ x,
                             const float* __restrict__ W1,
                             float* __restrict__ h1, int N) {
    const int lane    = threadIdx.x & 31;
    const int wave    = threadIdx.x >> 5;
    const int rowBase = blockIdx.x * 128 + wave * 16;
    const int m       = lane & 15;        // A: row M / B: col N within tile
    const int kOff    = (lane >> 4) * 2;  // lanes 16-31 carry K+2,K+3

    int row = rowBase + m;
    if (row >= N) row = N - 1;            // clamp loads; stores guarded below
    const float* xrow = x + (size_t)row * 128;

    v8f acc0 = {};   // cols 0..15
    v8f acc1 = {};   // cols 16..31

    for (int k0 = 0; k0 < 128; k0 += 4) {
        // A fragment: 16x4 f32, 2 VGPRs per lane (8-byte aligned load)
        v2f a = *(const v2f*)(xrow + k0 + kOff);
        // B fragments: 4x16 f32, W1 row-major [128][32]
        const int kr = (k0 + kOff) * 32 + m;
        v2f b0, b1;
        b0.x = W1[kr];      b0.y = W1[kr + 32];   // n0 = 0
        b1.x = W1[kr + 16]; b1.y = W1[kr + 48];   // n0 = 16
        acc0 = __builtin_amdgcn_wmma_f32_16x16x4_f32(false, a, false, b0,
                                                     (short)0, acc0, false, false);
        acc1 = __builtin_amdgcn_wmma_f32_16x16x4_f32(false, a, false, b1,
                                                     (short)0, acc1, false, false);
    }

    // D layout: VGPR r, lanes 0-15 -> M=r ; lanes 16-31 -> M=8+r ; N = lane&15
    const int mHi = (lane >> 4) * 8;
    for (int r = 0; r < 8; ++r) {
        int node = rowBase + mHi + r;
        if (node < N) {
            h1[(size_t)node * 32 + m]      = acc0[r];
            h1[(size_t)node * 32 + 16 + m] = acc1[r];
        }
    }
}

// ---------- per-node attention coefficients, layer 1 (2 heads x 16 ch) ----------
__global__ void k_att1(const float* __restrict__ h1,
                       const float* __restrict__ att_src,
                       const float* __restrict__ att_dst,
                       float* __restrict__ a_src, float* __restrict__ a_dst, int N) {
    int n = blockIdx.x * blockDim.x + threadIdx.x;
    if (n >= N) return;
    const float* hr = h1 + (size_t)n * 32;
    float s0 = 0.f, s1 = 0.f, d0 = 0.f, d1 = 0.f;
    for (int c = 0; c < 16; ++c) {
        float v0 = hr[c], v1 = hr[16 + c];
        s0 += v0 * att_src[c];      s1 += v1 * att_src[16 + c];
        d0 += v0 * att_dst[c];      d1 += v1 * att_dst[16 + c];
    }
    a_src[n * 2] = s0;  a_src[n * 2 + 1] = s1;
    a_dst[n * 2] = d0;  a_dst[n * 2 + 1] = d1;
}

// edge fetch: edge_index is [2,E] flat; edges >= E are self-loops
__device__ __forceinline__ void edge_get(const int* __restrict__ ei, long long E,
                                         long long e, int& s, int& d) {
    if (e < E) { s = ei[e]; d = ei[E + e]; }
    else       { s = d = (int)(e - E); }
}

// ---------- pass 1: scatter-max of leaky-relu logits ----------
template <int H>
__global__ void k_edge_max(const int* __restrict__ ei, long long E, long long Etot,
                           const float* __restrict__ a_src,
                           const float* __restrict__ a_dst,
                           unsigned* __restrict__ mflip) {
    long long e = (long long)blockIdx.x * blockDim.x + threadIdx.x;
    if (e >= Etot) return;
    if (e + 4096 < E) {
        __builtin_prefetch(&ei[e + 4096], 0, 0);
        __builtin_prefetch(&ei[E + e + 4096], 0, 0);
    }
    int s, d; edge_get(ei, E, e, s, d);
    for (int h = 0; h < H; ++h) {
        float v = a_src[s * H + h] + a_dst[d * H + h];
        v = v > 0.f ? v : NEG_SLOPE * v;
        atomicMax(&mflip[d * H + h], flipF(v));
    }
}

// ---------- pass 2: scatter-sum of exp(alpha - max) ----------
template <int H>
__global__ void k_edge_expsum(const int* __restrict__ ei, long long E, long long Etot,
                              const float* __restrict__ a_src,
                              const float* __restrict__ a_dst,
                              const unsigned* __restrict__ mflip,
                              float* __restrict__ ssum) {
    long long e = (long long)blockIdx.x * blockDim.x + threadIdx.x;
    if (e >= Etot) return;
    if (e + 4096 < E) {
        __builtin_prefetch(&ei[e + 4096], 0, 0);
        __builtin_prefetch(&ei[E + e + 4096], 0, 0);
    }
    int s, d; edge_get(ei, E, e, s, d);
    for (int h = 0; h < H; ++h) {
        float v = a_src[s * H + h] + a_dst[d * H + h];
        v = v > 0.f ? v : NEG_SLOPE * v;
        float ex = expf(v - unflipF(mflip[d * H + h]));
        atomicAdd(&ssum[d * H + h], ex);
    }
}

// ---------- pass 3, layer 1: weighted scatter-add of 32-channel features ----------
__global__ void k_edge_agg1(const int* __restrict__ ei, long long E, long long Etot,
                            const float* __restrict__ a_src,
                            const float* __restrict__ a_dst,
                            const unsigned* __restrict__ mflip,
                            const float* __restrict__ ssum,
                            const float* __restrict__ h1,
                            float* __restrict__ agg) {
    long long e = (long long)blockIdx.x * blockDim.x + threadIdx.x;
    if (e >= Etot) return;
    if (e + 4096 < E) {
        __builtin_prefetch(&ei[e + 4096], 0, 0);
        __builtin_prefetch(&ei[E + e + 4096], 0, 0);
    }
    int s, d; edge_get(ei, E, e, s, d);
    float w[2];
    for (int h = 0; h < 2; ++h) {
        float v = a_src[s * 2 + h] + a_dst[d * 2 + h];
        v = v > 0.f ? v : NEG_SLOPE * v;
        w[h] = expf(v - unflipF(mflip[d * 2 + h])) / (ssum[d * 2 + h] + 1e-16f);
    }
    const float* hs = h1 + (size_t)s * 32;
    float* od = agg + (size_t)d * 32;
    for (int c = 0; c < 32; ++c)
        atomicAdd(&od[c], w[c >> 4] * hs[c]);
}

// ---------- finalize layer1 (bias+ELU), project to layer2 feats + att coeffs ----------
__global__ void k_layer2_proj(const float* __restrict__ agg1,
                              const float* __restrict__ b1,
                              const float* __restrict__ W2,
                              const float* __restrict__ att_src2,
                              const float* __restrict__ att_dst2,
                              float* __restrict__ h2,
                              float* __restrict__ a_src2,
                              float* __restrict__ a_dst2, int N) {
    int n = blockIdx.x * blockDim.x + threadIdx.x;
    if (n >= N) return;
    const float* ar = agg1 + (size_t)n * 32;
    float o0 = 0.f, o1 = 0.f;
    for (int c = 0; c < 32; ++c) {
        float v = ar[c] + b1[c];
        v = v > 0.f ? v : expm1f(v);          // ELU (alpha=1)
        o0 += v * W2[c * 2];
        o1 += v * W2[c * 2 + 1];
    }
    h2[n * 2] = o0;  h2[n * 2 + 1] = o1;
    a_src2[n] = o0 * att_src2[0] + o1 * att_src2[1];
    a_dst2[n] = o0 * att_dst2[0] + o1 * att_dst2[1];
}

// ---------- pass 3, layer 2: weighted scatter-add of 2-channel features ----------
__global__ void k_edge_agg2(const int* __restrict__ ei, long long E, long long Etot,
                            const float* __restrict__ a_src,
                            const float* __restrict__ a_dst,
                            const unsigned* __restrict__ mflip,
                            const float* __restrict__ ssum,
                            const float* __restrict__ h2,
                            float* __restrict__ agg) {
    long long e = (long long)blockIdx.x * blockDim.x + threadIdx.x;
    if (e >= Etot) return;
    if (e + 4096 < E) {
        __builtin_prefetch(&ei[e + 4096], 0, 0);
        __builtin_prefetch(&ei[E + e + 4096], 0, 0);
    }
    int s, d; edge_get(ei, E, e, s, d);
    float v = a_src[s] + a_dst[d];
    v = v > 0.f ? v : NEG_SLOPE * v;
    float w = expf(v - unflipF(mflip[d])) / (ssum[d] + 1e-16f);
    atomicAdd(&agg[d * 2],     w * h2[s * 2]);
    atomicAdd(&agg[d * 2 + 1], w * h2[s * 2 + 1]);
}

// ---------- final: bias + log_softmax over 2 classes ----------
__global__ void k_logsoftmax(const float* __restrict__ agg2,
                             const float* __restrict__ b2,
                             float* __restrict__ out, int N) {
    int n = blockIdx.x * blockDim.x + threadIdx.x;
    if (n >= N) return;
    float o0 = agg2[n * 2] + b2[0];
    float o1 = agg2[n * 2 + 1] + b2[1];
    float mx = fmaxf(o0, o1);
    float lse = mx + logf(expf(o0 - mx) + expf(o1 - mx));
    out[n * 2]     = o0 - lse;
    out[n * 2 + 1] = o1 - lse;
}

extern "C" void kernel_launch(void* const* d_in, const int* in_sizes, int n_in,
                              void* d_out, int out_size, void* d_ws, size_t ws_size,
                              hipStream_t stream) {
    const float* x        = (const float*)d_in[0];
    const int*   ei       = (const int*)  d_in[1];
    const float* W1       = (const float*)d_in[2];
    const float* att_src1 = (const float*)d_in[3];
    const float* att_dst1 = (const float*)d_in[4];
    const float* b1       = (const float*)d_in[5];
    const float* W2       = (const float*)d_in[6];
    const float* att_src2 = (const float*)d_in[7];
    const float* att_dst2 = (const float*)d_in[8];
    const float* b2       = (const float*)d_in[9];

    const int       N    = in_sizes[0] / 128;
    const long long E    = in_sizes[1] / 2;
    const long long Etot = E + (long long)N;

    // ---- workspace layout (floats) ----
    float*    ws     = (float*)d_ws;
    float*    h1     = ws;                               // 32N
    float*    a_src1 = h1     + (size_t)N * 32;          // 2N
    float*    a_dst1 = a_src1 + (size_t)N * 2;           // 2N
    float*    h2     = a_dst1 + (size_t)N * 2;           // 2N
    float*    a_src2 = h2     + (size_t)N * 2;           // N
    float*    a_dst2 = a_src2 + (size_t)N;               // N
    float*    zbase  = a_dst2 + (size_t)N;               // zeroed region (40N):
    unsigned* m1     = (unsigned*)zbase;                 //   2N (flipped-float max)
    float*    s1     = zbase  + (size_t)N * 2;           //   2N
    float*    agg1   = s1     + (size_t)N * 2;           //   32N
    unsigned* m2     = (unsigned*)(agg1 + (size_t)N*32); //   N
    float*    s2     = agg1   + (size_t)N * 33;          //   N
    float*    agg2   = s2     + (size_t)N;               //   2N

    const size_t zcount = (size_t)N * 40;
    k_zero<<<dim3((unsigned)((zcount + 255) / 256)), 256, 0, stream>>>(zbase, zcount);

    // layer 1
    k_gemm1_wmma<<<dim3((N + 127) / 128), 256, 0, stream>>>(x, W1, h1, N);
    k_att1<<<dim3((N + 255) / 256), 256, 0, stream>>>(h1, att_src1, att_dst1,
                                                      a_src1, a_dst1, N);
    dim3 egrid((unsigned)((Etot + 255) / 256));
    k_edge_max<2>   <<<egrid, 256, 0, stream>>>(ei, E, Etot, a_src1, a_dst1, m1);
    k_edge_expsum<2><<<egrid, 256, 0, stream>>>(ei, E, Etot, a_src1, a_dst1, m1, s1);
    k_edge_agg1     <<<egrid, 256, 0, stream>>>(ei, E, Etot, a_src1, a_dst1, m1, s1,
                                                h1, agg1);
    // layer 2
    k_layer2_proj<<<dim3((N + 255) / 256), 256, 0, stream>>>(agg1, b1, W2,
                                                             att_src2, att_dst2,
                                                             h2, a_src2, a_dst2, N);
    k_edge_max<1>   <<<egrid, 256, 0, stream>>>(ei, E, Etot, a_src2, a_dst2, m2);
    k_edge_expsum<1><<<egrid, 256, 0, stream>>>(ei, E, Etot, a_src2, a_dst2, m2, s2);
    k_edge_agg2     <<<egrid, 256, 0, stream>>>(ei, E, Etot, a_src2, a_dst2, m2, s2,
                                                h2, agg2);
    k_logsoftmax<<<dim3((N + 255) / 256), 256, 0, stream>>>(agg2, b2, (float*)d_out, N);
}